// GSA_74045236183284
// MI455X (gfx1250) — compile-verified
//
#include <hip/hip_runtime.h>
#include <math.h>

#define Bdim 8
#define Cdim 128
#define Ndim 16384
#define Kdim 8
#define CN (Cdim*Ndim)

#define WSTR 130   // LDS row stride for conv weights (even -> 8B-aligned pairs)
#define GSTR 72    // LDS row stride for activations (2*72 % 64 == 16 -> half-waves hit disjoint banks)

typedef float v2f __attribute__((ext_vector_type(2)));
typedef float v8f __attribute__((ext_vector_type(8)));

__device__ __forceinline__ float gelu_exact(float x){
    return 0.5f * x * (1.0f + erff(x * 0.7071067811865475f));
}

// ---------------- zero scratch ----------------
__global__ void k_zero(float* __restrict__ p, int n){
    int i = blockIdx.x*blockDim.x + threadIdx.x;
    for (; i < n; i += gridDim.x*blockDim.x) p[i] = 0.f;
}

// ---------------- 1) bucketize tanh(logits); count per (b,k) ----------------
__global__ void k_bucket(const float* __restrict__ logits, int* __restrict__ bucket,
                         float* __restrict__ counts){
    __shared__ float lcnt[Kdim];
    int b = blockIdx.y;
    int n = blockIdx.x*256 + threadIdx.x;
    if (threadIdx.x < Kdim) lcnt[threadIdx.x] = 0.f;
    __syncthreads();
    float w = tanhf(logits[b*Ndim + n]);
    // bins: (lows[k], lows[k]+0.25], lows[k] = -1 + 0.25k  ->  idx = ceil((w+1)*4)-1
    int idx = (int)ceilf((w + 1.0f) * 4.0f) - 1;
    idx = idx < 0 ? 0 : (idx > 7 ? 7 : idx);
    if (w == 0.0f) idx = -1;           // masks exclude w == 0
    bucket[b*Ndim + n] = idx;
    if (idx >= 0) atomicAdd(&lcnt[idx], 1.0f);
    __syncthreads();
    if (threadIdx.x < Kdim) atomicAdd(&counts[b*Kdim + threadIdx.x], lcnt[threadIdx.x]);
}

// ---------------- 2) pooled[b,c,k] = sum_n x[b,c,n] * mask ----------------
__global__ void __launch_bounds__(256) k_pool(const float* __restrict__ x,
                                              const int* __restrict__ bucket,
                                              float* __restrict__ pooled){
    __shared__ float lacc[Cdim*Kdim];
    int b = blockIdx.y;
    int nb = blockIdx.x*1024;
    int tid = threadIdx.x;
    for (int i = tid; i < Cdim*Kdim; i += 256) lacc[i] = 0.f;
    int bk[4];
    #pragma unroll
    for (int j = 0; j < 4; j++) bk[j] = bucket[b*Ndim + nb + tid + j*256];
    __syncthreads();
    for (int c = 0; c < Cdim; c++){
        const float* xp = x + (size_t)b*CN + (size_t)c*Ndim + nb + tid;
        #pragma unroll
        for (int j = 0; j < 4; j++)
            if (bk[j] >= 0) atomicAdd(&lacc[c*Kdim + bk[j]], xp[j*256]);
    }
    __syncthreads();
    for (int i = tid; i < Cdim*Kdim; i += 256)
        atomicAdd(&pooled[b*Cdim*Kdim + i], lacc[i]);
}

// ---------------- 3) feature = pooled / max(nums,1) ----------------
__global__ void k_featinit(const float* __restrict__ pooled, const float* __restrict__ counts,
                           float* __restrict__ feat){
    int i = blockIdx.x*256 + threadIdx.x;
    if (i >= Bdim*Cdim*Kdim) return;
    int b = i / (Cdim*Kdim);
    int kk = i % Kdim;
    float cnt = counts[b*Kdim + kk];
    if (cnt == 0.f) cnt = 1.f;
    feat[i] = pooled[i] / cnt;
}

// ---------------- 4) 3-layer attention over tiny [128 x 8] feature, per b ----------------
__global__ void __launch_bounds__(256) k_attn(float* __restrict__ feat,
        const float* Wq1, const float* Wk1, const float* Wv1,
        const float* Wq2, const float* Wk2, const float* Wv2,
        const float* Wq3, const float* Wk3, const float* Wv3,
        const float* __restrict__ ln_w, const float* __restrict__ ln_b){
    __shared__ float f [Cdim*Kdim];
    __shared__ float qq[Cdim*Kdim];
    __shared__ float kb[Cdim*Kdim];
    __shared__ float vb[Cdim*Kdim];
    __shared__ float ob[Cdim*Kdim];
    __shared__ float sc[32*Cdim];
    __shared__ float mu8[Kdim], vr8[Kdim];
    int b = blockIdx.x, tid = threadIdx.x;
    const float* Wq[3] = {Wq1, Wq2, Wq3};
    const float* Wk[3] = {Wk1, Wk2, Wk3};
    const float* Wv[3] = {Wv1, Wv2, Wv3};
    const float inv_temp = 0.08838834764831845f;   // 1/sqrt(128)
    for (int i = tid; i < Cdim*Kdim; i += 256) f[i] = feat[b*Cdim*Kdim + i];
    __syncthreads();
    for (int layer = 0; layer < 3; layer++){
        const float* wq = Wq[layer];
        const float* wk = Wk[layer];
        const float* wv = Wv[layer];
        // projections: 1024 outputs each, dot over 128
        for (int i = tid; i < Cdim*Kdim; i += 256){
            int c = i >> 3, n = i & 7;
            float aq = 0.f, ak = 0.f, av = 0.f;
            for (int d = 0; d < Cdim; d++){
                float fv = f[d*Kdim + n];
                aq += wq[c*Cdim + d] * fv;
                ak += wk[c*Cdim + d] * fv;
                av += wv[c*Cdim + d] * fv;
            }
            qq[i] = aq; kb[i] = ak; vb[i] = av;
        }
        __syncthreads();
        // attention, 32 rows at a time
        for (int r0 = 0; r0 < Cdim; r0 += 32){
            #pragma unroll
            for (int j = 0; j < 16; j++){
                int si = tid + 256*j;          // si = i*128 + d
                int i = si >> 7, d = si & 127;
                float s = 0.f;
                #pragma unroll
                for (int n = 0; n < Kdim; n++) s += qq[(r0+i)*Kdim + n] * kb[d*Kdim + n];
                sc[si] = s * inv_temp;
            }
            __syncthreads();
            if (tid < 32){
                float m = -1e30f;
                for (int d = 0; d < Cdim; d++) m = fmaxf(m, sc[tid*Cdim + d]);
                float s = 0.f;
                for (int d = 0; d < Cdim; d++){ float e = expf(sc[tid*Cdim + d] - m); sc[tid*Cdim + d] = e; s += e; }
                float inv = 1.0f / s;
                for (int d = 0; d < Cdim; d++) sc[tid*Cdim + d] *= inv;
            }
            __syncthreads();
            {
                int i = tid >> 3, n = tid & 7;
                float s = 0.f;
                for (int d = 0; d < Cdim; d++) s += sc[i*Cdim + d] * vb[d*Kdim + n];
                ob[(r0+i)*Kdim + n] = s;
            }
            __syncthreads();
        }
        // residual + layernorm over c (per n)
        if (tid < Kdim){
            float s = 0.f;
            for (int c = 0; c < Cdim; c++) s += ob[c*Kdim + tid] + f[c*Kdim + tid];
            float mu = s * (1.0f/Cdim);
            float s2 = 0.f;
            for (int c = 0; c < Cdim; c++){
                float v = ob[c*Kdim + tid] + f[c*Kdim + tid] - mu;
                s2 += v*v;
            }
            mu8[tid] = mu; vr8[tid] = s2 * (1.0f/Cdim);
        }
        __syncthreads();
        for (int i = tid; i < Cdim*Kdim; i += 256){
            int c = i >> 3, n = i & 7;
            float v = ob[i] + f[i];
            f[i] = (v - mu8[n]) * rsqrtf(vr8[n] + 1e-6f) * ln_w[c] + ln_b[c];
        }
        __syncthreads();
    }
    for (int i = tid; i < Cdim*Kdim; i += 256) feat[b*Cdim*Kdim + i] = f[i];
}

// ---------------- 5) per-(b,c) sum / sumsq of recon over n ----------------
__global__ void __launch_bounds__(256) k_stats(const float* __restrict__ x,
        const int* __restrict__ bucket, const float* __restrict__ feat,
        float* __restrict__ sums, float* __restrict__ sumsq){
    __shared__ float fl[Cdim*Kdim];
    __shared__ float red[8], red2[8];
    int b = blockIdx.y, nb = blockIdx.x*1024, tid = threadIdx.x;
    for (int i = tid; i < Cdim*Kdim; i += 256) fl[i] = feat[b*Cdim*Kdim + i];
    int bk[4];
    #pragma unroll
    for (int j = 0; j < 4; j++) bk[j] = bucket[b*Ndim + nb + tid + j*256];
    __syncthreads();
    int lane = tid & 31, wid = tid >> 5;
    for (int c = 0; c < Cdim; c++){
        const float* xp = x + (size_t)b*CN + (size_t)c*Ndim + nb + tid;
        float s = 0.f, s2 = 0.f;
        #pragma unroll
        for (int j = 0; j < 4; j++){
            float v = 0.f;
            if (bk[j] >= 0) v = xp[j*256] + fl[c*Kdim + bk[j]];  // recon (total==1 else 0)
            s += v; s2 += v*v;
        }
        for (int off = 16; off > 0; off >>= 1){
            s  += __shfl_down(s,  off, 32);
            s2 += __shfl_down(s2, off, 32);
        }
        if (lane == 0){ red[wid] = s; red2[wid] = s2; }
        __syncthreads();
        if (tid == 0){
            float ts = 0.f, ts2 = 0.f;
            for (int w = 0; w < 8; w++){ ts += red[w]; ts2 += red2[w]; }
            atomicAdd(&sums[b*Cdim + c], ts);
            atomicAdd(&sumsq[b*Cdim + c], ts2);
        }
        __syncthreads();
    }
}

// ---------------- 6) collapse instance-norm + batch-norm into affine A,B ----------------
// h1 = (recon - mu_i)*rsqrt(var_i+1e-5)  has mean 0 / var var_i/(var_i+eps) over n,
// so mu_b == 0 exactly and var_b[c] = mean_b[ var_i/(var_i+1e-5) ].
__global__ void k_affine(const float* __restrict__ sums, const float* __restrict__ sumsq,
                         const float* __restrict__ bn_w, const float* __restrict__ bn_b,
                         float* __restrict__ Aarr, float* __restrict__ Brr){
    int c = threadIdx.x;               // 1 block, 128 threads
    float mu[Bdim], ivar[Bdim];
    float vb_acc = 0.f;
    for (int b = 0; b < Bdim; b++){
        float m   = sums [b*Cdim + c] * (1.0f/Ndim);
        float var = sumsq[b*Cdim + c] * (1.0f/Ndim) - m*m;
        var = fmaxf(var, 0.f);
        mu[b] = m;
        ivar[b] = rsqrtf(var + 1e-5f);
        vb_acc += var / (var + 1e-5f);
    }
    float ivb = rsqrtf(vb_acc * (1.0f/Bdim) + 1e-5f);
    for (int b = 0; b < Bdim; b++){
        float A = ivar[b] * ivb * bn_w[c];
        Aarr[b*Cdim + c] = A;
        Brr [b*Cdim + c] = bn_b[c] - mu[b]*A;
    }
}

// ---------------- 7) fused recon -> affine -> gelu -> 128x128 conv GEMM (f32 WMMA) ----
// Per 128(o) x 64(n) block: wave w owns o-rows [16w,16w+16); its A fragments are
// loaded to registers ONCE, then each 16-col n-tile bulk-loads B fragments and runs
// a back-to-back dual-accumulator WMMA chain (one ds wait per tile, not per wmma).
__global__ void __launch_bounds__(256) k_conv(const float* __restrict__ x,
        const int* __restrict__ bucket, const float* __restrict__ feat,
        const float* __restrict__ Aarr, const float* __restrict__ Brr,
        const float* __restrict__ conv0_w, const float* __restrict__ conv0_b,
        float* __restrict__ out){
    extern __shared__ float smem[];
    float* Wl = smem;                  // 128 * WSTR
    float* gl = Wl + Cdim*WSTR;        // 128 * GSTR
    float* fl = gl + Cdim*GSTR;        // 1024
    float* Av = fl + Cdim*Kdim;        // 128
    float* Bv = Av + Cdim;             // 128
    int b = blockIdx.y;
    int nBase = blockIdx.x * 64;
    int tid = threadIdx.x;

    for (int i = tid; i < Cdim*Cdim; i += 256){
        int o = i >> 7, c = i & 127;
        Wl[o*WSTR + c] = conv0_w[i];
    }
    for (int i = tid; i < Cdim*Kdim; i += 256) fl[i] = feat[b*Cdim*Kdim + i];
    if (tid < Cdim){ Av[tid] = Aarr[b*Cdim + tid]; Bv[tid] = Brr[b*Cdim + tid]; }

    // g[c][n] = gelu(recon * A + B), coalesced in n
    int nl = tid & 63;
    int c0 = tid >> 6;
    int n  = nBase + nl;
    int bk = bucket[b*Ndim + n];
    __builtin_prefetch(x + (size_t)b*CN + (size_t)c0*Ndim + n, 0, 1);
    for (int c = c0; c < Cdim; c += 4){
        float v = 0.f;
        if (bk >= 0) v = x[(size_t)b*CN + (size_t)c*Ndim + n] + fl[c*Kdim + bk];
        gl[c*GSTR + nl] = gelu_exact(v*Av[c] + Bv[c]);
    }
    __syncthreads();

    int lane = tid & 31, wid = tid >> 5;
    int lo = lane & 15, hi = lane >> 4;
    int o0 = wid * 16;                 // each wave owns one 16-row o-strip

    // A 16x4 fragments: lane(0-15)=row M; vgpr0 -> K=2*hi, vgpr1 -> K=2*hi+1
    v2f afr[32];
    #pragma unroll
    for (int k4 = 0; k4 < 32; k4++){
        const float* ap = &Wl[(o0 + lo)*WSTR + 4*k4 + 2*hi];
        afr[k4].x = ap[0];
        afr[k4].y = ap[1];
    }

    #pragma unroll
    for (int nt = 0; nt < 4; nt++){
        int n0 = nt * 16;
        // B 4x16 fragments: lane(0-15)=col N, same K striping
        v2f bfr[32];
        #pragma unroll
        for (int k4 = 0; k4 < 32; k4++){
            bfr[k4].x = gl[(4*k4 + 2*hi    )*GSTR + n0 + lo];
            bfr[k4].y = gl[(4*k4 + 2*hi + 1)*GSTR + n0 + lo];
        }
        v8f acc0 = {}, acc1 = {};
        #pragma unroll
        for (int k4 = 0; k4 < 32; k4 += 2){
            acc0 = __builtin_amdgcn_wmma_f32_16x16x4_f32(
                false, afr[k4],   false, bfr[k4],   (short)0, acc0, false, false);
            acc1 = __builtin_amdgcn_wmma_f32_16x16x4_f32(
                false, afr[k4+1], false, bfr[k4+1], (short)0, acc1, false, false);
        }
        #pragma unroll
        for (int r = 0; r < 8; r++){
            int row = o0 + r + 8*hi;          // C/D: vgpr r, halves hold M=r / M=r+8
            int col = nBase + n0 + lo;
            out[(size_t)b*CN + (size_t)row*Ndim + col] = acc0[r] + acc1[r] + conv0_b[row];
        }
    }
}

// ---------------- 8) sq[b,o] = mean_n conv ----------------
__global__ void k_squeeze(const float* __restrict__ out, float* __restrict__ sq){
    __shared__ float red[8];
    int bo = blockIdx.x;                   // b*128 + o
    int tid = threadIdx.x;
    const float* p = out + (size_t)bo * Ndim;
    float s = 0.f;
    for (int n = tid; n < Ndim; n += 256) s += p[n];
    for (int off = 16; off > 0; off >>= 1) s += __shfl_down(s, off, 32);
    int lane = tid & 31, wid = tid >> 5;
    if (lane == 0) red[wid] = s;
    __syncthreads();
    if (tid == 0){
        float t = 0.f;
        for (int w = 0; w < 8; w++) t += red[w];
        sq[bo] = t * (1.0f/Ndim);
    }
}

// ---------------- 9) squeeze-excite MLP ----------------
__global__ void k_fc(const float* __restrict__ sq,
                     const float* __restrict__ fc1_w, const float* __restrict__ fc1_b,
                     const float* __restrict__ fc2_w, const float* __restrict__ fc2_b,
                     float* __restrict__ f2g){
    __shared__ float s[Cdim], f1[64];
    int b = blockIdx.x, tid = threadIdx.x;   // 128 threads
    s[tid] = sq[b*Cdim + tid];
    __syncthreads();
    if (tid < 64){
        float a = fc1_b[tid];
        for (int c = 0; c < Cdim; c++) a += s[c] * fc1_w[tid*Cdim + c];
        f1[tid] = gelu_exact(a);
    }
    __syncthreads();
    float a = fc2_b[tid];
    for (int j = 0; j < 64; j++) a += f1[j] * fc2_w[tid*64 + j];
    f2g[b*Cdim + tid] = 1.0f / (1.0f + expf(-a));
}

// ---------------- 10) out *= f2 ----------------
__global__ void k_scale(float* __restrict__ out, const float* __restrict__ f2g){
    int i = blockIdx.x*256 + threadIdx.x;
    const int total = Bdim*CN;
    for (; i < total; i += gridDim.x*256){
        int bc = i >> 14;                  // i / N, N = 2^14 -> b*128 + c
        out[i] *= f2g[bc];
    }
}

extern "C" void kernel_launch(void* const* d_in, const int* in_sizes, int n_in,
                              void* d_out, int out_size, void* d_ws, size_t ws_size,
                              hipStream_t stream){
    (void)in_sizes; (void)n_in; (void)out_size; (void)ws_size;
    const float* x       = (const float*)d_in[0];
    const float* logits  = (const float*)d_in[1];
    const float* Wq1     = (const float*)d_in[2];
    const float* Wk1     = (const float*)d_in[3];
    const float* Wv1     = (const float*)d_in[4];
    const float* Wq2     = (const float*)d_in[5];
    const float* Wk2     = (const float*)d_in[6];
    const float* Wv2     = (const float*)d_in[7];
    const float* Wq3     = (const float*)d_in[8];
    const float* Wk3     = (const float*)d_in[9];
    const float* Wv3     = (const float*)d_in[10];
    const float* ln_w    = (const float*)d_in[11];
    const float* ln_b    = (const float*)d_in[12];
    const float* conv0_w = (const float*)d_in[13];
    const float* conv0_b = (const float*)d_in[14];
    const float* bn_w    = (const float*)d_in[15];
    const float* bn_b    = (const float*)d_in[16];
    const float* fc1_w   = (const float*)d_in[17];
    const float* fc1_b   = (const float*)d_in[18];
    const float* fc2_w   = (const float*)d_in[19];
    const float* fc2_b   = (const float*)d_in[20];
    float* out = (float*)d_out;

    float* wsf    = (float*)d_ws;
    int*   bucket = (int*)wsf;                         // B*N ints
    float* counts = wsf + Bdim*Ndim;                   // B*K
    float* pooled = counts + Bdim*Kdim;                // B*C*K
    float* sums   = pooled + Bdim*Cdim*Kdim;           // B*C
    float* sumsq  = sums   + Bdim*Cdim;                // B*C
    float* feat   = sumsq  + Bdim*Cdim;                // B*C*K
    float* Aarr   = feat   + Bdim*Cdim*Kdim;           // B*C
    float* Brr    = Aarr   + Bdim*Cdim;                // B*C
    float* sq     = Brr    + Bdim*Cdim;                // B*C
    float* f2g    = sq     + Bdim*Cdim;                // B*C

    int nzero = Bdim*Kdim + Bdim*Cdim*Kdim + 2*Bdim*Cdim; // counts..sumsq (contiguous)
    k_zero    <<<64, 256, 0, stream>>>(counts, nzero);
    k_bucket  <<<dim3(Ndim/256,  Bdim), 256, 0, stream>>>(logits, bucket, counts);
    k_pool    <<<dim3(Ndim/1024, Bdim), 256, 0, stream>>>(x, bucket, pooled);
    k_featinit<<<(Bdim*Cdim*Kdim + 255)/256, 256, 0, stream>>>(pooled, counts, feat);
    k_attn    <<<Bdim, 256, 0, stream>>>(feat, Wq1,Wk1,Wv1, Wq2,Wk2,Wv2, Wq3,Wk3,Wv3, ln_w, ln_b);
    k_stats   <<<dim3(Ndim/1024, Bdim), 256, 0, stream>>>(x, bucket, feat, sums, sumsq);
    k_affine  <<<1, 128, 0, stream>>>(sums, sumsq, bn_w, bn_b, Aarr, Brr);
    size_t shConv = (size_t)(Cdim*WSTR + Cdim*GSTR + Cdim*Kdim + 2*Cdim) * sizeof(float);
    k_conv    <<<dim3(Ndim/64, Bdim), 256, shConv, stream>>>(x, bucket, feat, Aarr, Brr,
                                                             conv0_w, conv0_b, out);
    k_squeeze <<<Bdim*Cdim, 256, 0, stream>>>(out, sq);
    k_fc      <<<Bdim, 128, 0, stream>>>(sq, fc1_w, fc1_b, fc2_w, fc2_b, f2g);
    k_scale   <<<4096, 256, 0, stream>>>(out, f2g);
}